// BaseTPREncoder_43911745634914
// MI455X (gfx1250) — compile-verified
//
#include <hip/hip_runtime.h>
#include <hip/hip_bf16.h>
#include <math.h>

#define N_BATCH  256
#define N_ROLES  32
#define DIMV     4096
#define N_FILLWT 256

typedef __attribute__((ext_vector_type(2))) float v2f;
typedef __attribute__((ext_vector_type(8))) float v8f;

// ---------------------------------------------------------------------------
// Kernel 1: per-row stable key-value bitonic sort.
// binding[j] = filler[argsort(role)[j]]  ==  fillers sorted by role key.
// 64-bit items = (orderable(role) << 32) | index  -> stable, exact argsort.
// One workgroup (256 threads, 8 waves) per (n,r) row; 48KB LDS.
// ---------------------------------------------------------------------------
__global__ __launch_bounds__(256) void sort_bind_kernel(const float* __restrict__ roles,
                                                        const float* __restrict__ fillers,
                                                        float* __restrict__ bindings) {
    __shared__ unsigned long long items[DIMV];  // 32 KB
    __shared__ float fil[DIMV];                 // 16 KB
    const int row = blockIdx.x;
    const float* rrow = roles   + (size_t)row * DIMV;
    const float* frow = fillers + (size_t)row * DIMV;
    for (int d = threadIdx.x; d < DIMV; d += 256) {
        unsigned u = __float_as_uint(rrow[d]);
        u = (u & 0x80000000u) ? ~u : (u | 0x80000000u);   // float -> orderable uint
        items[d] = ((unsigned long long)u << 32) | (unsigned)d;
        fil[d] = frow[d];
    }
    __syncthreads();
    for (int k = 2; k <= DIMV; k <<= 1) {
        for (int j = k >> 1; j > 0; j >>= 1) {
            for (int i = threadIdx.x; i < DIMV; i += 256) {
                int ixj = i ^ j;
                if (ixj > i) {
                    unsigned long long a = items[i];
                    unsigned long long b = items[ixj];
                    bool doSwap = ((i & k) == 0) ? (a > b) : (a < b);
                    if (doSwap) { items[i] = b; items[ixj] = a; }
                }
            }
            __syncthreads();
        }
    }
    float* brow = bindings + (size_t)row * DIMV;
    for (int d = threadIdx.x; d < DIMV; d += 256) {
        brow[d] = fil[(unsigned)(items[d] & 0xffffffffu)];
    }
}

// ---------------------------------------------------------------------------
// Kernel 2: z_rep[n,d] = sum_r bindings[n,r,d]
// ---------------------------------------------------------------------------
__global__ __launch_bounds__(256) void zrep_kernel(const float* __restrict__ bindings,
                                                   float* __restrict__ z) {
    size_t idx = (size_t)blockIdx.x * 256 + threadIdx.x;   // over N*D
    int n = (int)(idx / DIMV);
    int d = (int)(idx % DIMV);
    const float* base = bindings + ((size_t)n * N_ROLES) * DIMV + d;
    float s = 0.f;
    for (int r = 0; r < N_ROLES; ++r) s += base[(size_t)r * DIMV];
    z[idx] = s;
}

// ---------------------------------------------------------------------------
// Kernel 3: Gram matrix G = W * W^T via V_WMMA_F32_16X16X4_F32 (full f32).
// One wave per 16x16 output tile. A-layout (32-bit 16x4, ISA 7.12.2):
//   lanes 0-15: M=lane, v0=K, v1=K+1 ; lanes 16-31: M=lane-16, v0=K+2, v1=K+3
// B (4x16) mirrors with N per lane. C/D: vgpr v -> M = v + 8*(lane>>4), N = lane&15.
// ---------------------------------------------------------------------------
__global__ __launch_bounds__(256) void gram_wmma_kernel(const float* __restrict__ W,
                                                        float* __restrict__ G, int n) {
    const int wave = blockIdx.x * 8 + (threadIdx.x >> 5);
    const int tilesPerSide = n >> 4;
    if (wave >= tilesPerSide * tilesPerSide) return;   // wave-uniform: EXEC all-1 for WMMA
    const int ti = wave / tilesPerSide;
    const int tj = wave % tilesPerSide;
    const int lane    = threadIdx.x & 31;
    const int laneLow = lane & 15;
    const int laneHi  = lane >> 4;
    const float* arow = W + (size_t)(ti * 16 + laneLow) * DIMV + 2 * laneHi;
    const float* brow = W + (size_t)(tj * 16 + laneLow) * DIMV + 2 * laneHi;
    v8f c = {};
    for (int k = 0; k < DIMV; k += 4) {
        v2f a, b;
        a.x = arow[k];     a.y = arow[k + 1];
        b.x = brow[k];     b.y = brow[k + 1];
        c = __builtin_amdgcn_wmma_f32_16x16x4_f32(false, a, false, b,
                                                  (short)0, c, false, false);
    }
    for (int v = 0; v < 8; ++v) {
        int gm = ti * 16 + laneHi * 8 + v;
        int gn = tj * 16 + laneLow;
        G[gm * n + gn] = c[v];
    }
}

// ---------------------------------------------------------------------------
// Kernel 4: penalty = sqrt( sum (I - G)^2 )   (single workgroup)
// ---------------------------------------------------------------------------
__global__ __launch_bounds__(256) void penalty_kernel(const float* __restrict__ G, int n,
                                                      float* __restrict__ out_pen) {
    __shared__ float red[256];
    const int tid = threadIdx.x;
    float s = 0.f;
    for (int idx = tid; idx < n * n; idx += 256) {
        int i = idx / n, j = idx % n;
        float d = ((i == j) ? 1.0f : 0.0f) - G[idx];
        s += d * d;
    }
    red[tid] = s;
    __syncthreads();
    for (int off = 128; off > 0; off >>= 1) {
        if (tid < off) red[tid] += red[tid + off];
        __syncthreads();
    }
    if (tid == 0) out_pen[0] = sqrtf(red[0]);
}

// ---------------------------------------------------------------------------
// Kernel 5: rank stat via Gaussian elimination w/ partial pivoting on G
// (rank(W^T) == rank(W) == rank(W W^T)). Destroys G. Single workgroup.
// stat = (n - rank) / (n - 1)
// ---------------------------------------------------------------------------
__global__ __launch_bounds__(256) void rank_kernel(float* __restrict__ G, int n,
                                                   float* __restrict__ out_rank) {
    __shared__ float smax[256];
    __shared__ int   sarg[256];
    __shared__ int   rankCount;
    __shared__ float tolSh;
    __shared__ int   pivSh;
    __shared__ int   okSh;
    const int tid = threadIdx.x;
    if (tid == 0) rankCount = 0;
    float lm = 0.f;
    for (int idx = tid; idx < n * n; idx += 256) lm = fmaxf(lm, fabsf(G[idx]));
    smax[tid] = lm;
    __syncthreads();
    for (int off = 128; off > 0; off >>= 1) {
        if (tid < off) smax[tid] = fmaxf(smax[tid], smax[tid + off]);
        __syncthreads();
    }
    if (tid == 0) tolSh = smax[0] * 1e-6f * (float)n;
    __syncthreads();
    const float tol = tolSh;
    for (int k = 0; k < n; ++k) {
        float m = 0.f; int arg = k;
        for (int i = k + tid; i < n; i += 256) {
            float v = fabsf(G[i * n + k]);
            if (v > m) { m = v; arg = i; }
        }
        smax[tid] = m; sarg[tid] = arg;
        __syncthreads();
        for (int off = 128; off > 0; off >>= 1) {
            if (tid < off && smax[tid + off] > smax[tid]) {
                smax[tid] = smax[tid + off]; sarg[tid] = sarg[tid + off];
            }
            __syncthreads();
        }
        if (tid == 0) {
            okSh  = (smax[0] > tol) ? 1 : 0;
            pivSh = sarg[0];
            if (okSh) rankCount++;
        }
        __syncthreads();
        if (okSh) {                 // uniform branch: barriers inside are safe
            const int p = pivSh;
            if (p != k) {
                for (int j2 = tid; j2 < n; j2 += 256) {
                    float t = G[k * n + j2];
                    G[k * n + j2] = G[p * n + j2];
                    G[p * n + j2] = t;
                }
            }
            __syncthreads();
            const float pk = G[k * n + k];
            for (int i = k + 1 + tid; i < n; i += 256) {
                const float f = G[i * n + k] / pk;
                for (int j2 = k; j2 < n; ++j2) G[i * n + j2] -= f * G[k * n + j2];
            }
        }
        __syncthreads();
    }
    if (tid == 0) out_rank[0] = (float)(n - rankCount) / (float)(n - 1);
}

// ---------------------------------------------------------------------------
// Kernel 6: encoder_loss = 0.01 * p_role + 0.01 * p_filler
// scalars layout: [0]=loss [1]=p_role [2]=p_filler [3]=role_rank [4]=filler_rank
// ---------------------------------------------------------------------------
__global__ void finalize_kernel(float* __restrict__ scalars) {
    if (threadIdx.x == 0)
        scalars[0] = 0.01f * scalars[1] + 0.01f * scalars[2];
}

extern "C" void kernel_launch(void* const* d_in, const int* in_sizes, int n_in,
                              void* d_out, int out_size, void* d_ws, size_t ws_size,
                              hipStream_t stream) {
    const float* roles   = (const float*)d_in[0];   // [256,32,4096]
    const float* fillers = (const float*)d_in[1];   // [256,32,4096]
    const float* roleW   = (const float*)d_in[2];   // [32,4096]
    const float* fillW   = (const float*)d_in[3];   // [256,4096]
    float* out = (float*)d_out;

    float* z_rep    = out;                                              // 1,048,576
    float* bindings = out + (size_t)N_BATCH * DIMV;                     // 33,554,432
    float* scalars  = bindings + (size_t)N_BATCH * N_ROLES * DIMV;      // 5 floats

    float* ws     = (float*)d_ws;
    float* G_fill = ws;                               // 256*256 floats
    float* G_role = ws + N_FILLWT * N_FILLWT;         //  32*32  floats

    // 1) bind via key-value sort (also the only argsort consumer)
    sort_bind_kernel<<<N_BATCH * N_ROLES, 256, 0, stream>>>(roles, fillers, bindings);
    // 2) superposition sum over roles
    zrep_kernel<<<(N_BATCH * DIMV) / 256, 256, 0, stream>>>(bindings, z_rep);

    // 3) role weight: Gram (WMMA f32), penalty, rank (destroys Gram)
    gram_wmma_kernel<<<1, 256, 0, stream>>>(roleW, G_role, N_ROLES);          // 4 tiles
    penalty_kernel<<<1, 256, 0, stream>>>(G_role, N_ROLES, scalars + 1);
    rank_kernel<<<1, 256, 0, stream>>>(G_role, N_ROLES, scalars + 3);

    // 4) filler weight: same pipeline
    gram_wmma_kernel<<<32, 256, 0, stream>>>(fillW, G_fill, N_FILLWT);        // 256 tiles
    penalty_kernel<<<1, 256, 0, stream>>>(G_fill, N_FILLWT, scalars + 2);
    rank_kernel<<<1, 256, 0, stream>>>(G_fill, N_FILLWT, scalars + 4);

    // 5) loss
    finalize_kernel<<<1, 32, 0, stream>>>(scalars);
}